// SparseRandomAttention_89472758710436
// MI455X (gfx1250) — compile-verified
//
#include <hip/hip_runtime.h>
#include <hip/hip_bf16.h>

#define N_NODES 4096
#define EMB     512
#define HEADS   8
#define HD      64
#define KSP     128

#define BM 128   // block tile rows
#define BN 16    // block tile cols (one d-slab)
#define BK 64    // staged k-chunk

typedef float v2f __attribute__((ext_vector_type(2)));
typedef float v4f __attribute__((ext_vector_type(4)));
typedef float v8f __attribute__((ext_vector_type(8)));

// ---------------------------------------------------------------------------
// Kernel A: C[n,d] = sum_c E[n,c] * W[d,c] + b[d]   (torch Linear, W.T form)
// One block = 128x16 output tile, 8 waves each own a 16x16 WMMA accumulator.
// f32 WMMA 16x16x4, K staged through padded LDS.
// ---------------------------------------------------------------------------
__global__ __launch_bounds__(256) void proj_gemm_kernel(
    const float* __restrict__ E,
    const float* __restrict__ Wq, const float* __restrict__ bq,
    const float* __restrict__ Wk, const float* __restrict__ bk,
    float* __restrict__ Qout, float* __restrict__ Kout)
{
    const float* W    = (blockIdx.z == 0) ? Wq : Wk;
    const float* bias = (blockIdx.z == 0) ? bq : bk;
    float*       Out  = (blockIdx.z == 0) ? Qout : Kout;

    __shared__ float Elds[BM][BK + 2];  // +2 pad: conflict-free strided ds_load_b64
    __shared__ float Wlds[BN][BK + 2];

    const int t    = threadIdx.x;       // 0..255
    const int wave = t >> 5;            // 0..7 -> 16-row slab
    const int lane = t & 31;
    const int ml   = lane & 15;         // A: m   B: n   D: column
    const int kh   = lane >> 4;         // selects K pair {0,1} vs {2,3}

    const int m0 = blockIdx.x * BM;
    const int d0 = blockIdx.y * BN;

    v8f acc = {};

    for (int c0 = 0; c0 < EMB; c0 += BK) {
        // Stage E chunk: 128 rows x 64 floats = 2048 float4, coalesced.
#pragma unroll
        for (int i = 0; i < 8; ++i) {
            int idx = t + 256 * i;      // float4 index within chunk
            int r   = idx >> 4;         // 16 float4 per row
            int c4  = idx & 15;
            v4f v = *(const v4f*)(E + (size_t)(m0 + r) * EMB + c0 + c4 * 4);
            *(v4f*)&Elds[r][c4 * 4] = v;
        }
        // Stage W chunk: 16 rows x 64 floats = 256 float4, one per thread.
        {
            int r  = t >> 4;
            int c4 = t & 15;
            v4f v = *(const v4f*)(W + (size_t)(d0 + r) * EMB + c0 + c4 * 4);
            *(v4f*)&Wlds[r][c4 * 4] = v;
        }
        __syncthreads();

        const int mrow = wave * 16;
#pragma unroll
        for (int cc = 0; cc < BK; cc += 4) {
            // A 16x4: lane(m=ml) vgpr j holds E[m][cc + 2*kh + j]
            v2f a = *(const v2f*)&Elds[mrow + ml][cc + 2 * kh];
            // B 4x16: B[k][n] = W[d0+n][c0+k]; lane(n=ml) vgpr j holds k=2*kh+j
            v2f b = *(const v2f*)&Wlds[ml][cc + 2 * kh];
            acc = __builtin_amdgcn_wmma_f32_16x16x4_f32(
                false, a, false, b, (short)0, acc, false, false);
        }
        __syncthreads();
    }

    // D layout: acc[i] = element (row = wave*16 + 8*kh + i, col = d0 + ml)
    const int   colg  = d0 + ml;
    const float bv    = bias[colg];
    const int   rbase = m0 + wave * 16 + 8 * kh;
#pragma unroll
    for (int i = 0; i < 8; ++i) {
        Out[(size_t)(rbase + i) * EMB + colg] = acc[i] + bv;
    }
}

// ---------------------------------------------------------------------------
// Kernel B: one block per (h, n) output row. Build the 4096-float row in LDS
// (zeros + 128 scattered scores, last-k-wins for duplicates), then stream it
// out with coalesced nontemporal b128 stores — each output line written once.
// ---------------------------------------------------------------------------
__global__ __launch_bounds__(128) void sparse_row_kernel(
    const float* __restrict__ Q, const float* __restrict__ Kmat,
    const int* __restrict__ sidx, float* __restrict__ out)
{
    __shared__ float row[N_NODES];   // 16 KB
    __shared__ float qs[HD];
    __shared__ int   il[KSP];

    const int t = threadIdx.x;       // 0..127, doubles as k
    const int b = blockIdx.x;        // h * N_NODES + n
    const int h = b >> 12;
    const int n = b & (N_NODES - 1);

    // Zero the LDS row: 128 threads x 8 float4.
    const v4f vz = {0.f, 0.f, 0.f, 0.f};
#pragma unroll
    for (int i = 0; i < 8; ++i)
        *(v4f*)&row[(t + 128 * i) * 4] = vz;

    if (t < HD) qs[t] = Q[(size_t)n * EMB + h * HD + t];
    il[t] = sidx[((size_t)h * N_NODES + n) * KSP + t];
    __syncthreads();

    // Score for key t: dot(q[h,n,:], K[idx, h*64 : h*64+64]) / sqrt(64)
    const int idx = il[t];
    const float* krow = Kmat + (size_t)idx * EMB + h * HD;
    float a = 0.f;
#pragma unroll
    for (int i = 0; i < 16; ++i) {
        v4f kv = *(const v4f*)(krow + i * 4);
        v4f qv = *(const v4f*)&qs[i * 4];
        a += kv.x * qv.x + kv.y * qv.y + kv.z * qv.z + kv.w * qv.w;
    }
    a *= 0.125f;

    // Deterministic duplicate resolution: only the last k with this index writes.
    bool last = true;
    for (int kk = t + 1; kk < KSP; ++kk)
        if (il[kk] == idx) { last = false; break; }
    if (last) row[idx] = a;
    __syncthreads();

    // Stream the finished row to HBM (write-once, nontemporal).
    float* orow = out + (size_t)b * N_NODES;
#pragma unroll
    for (int i = 0; i < 8; ++i) {
        v4f v = *(const v4f*)&row[(t + 128 * i) * 4];
        __builtin_nontemporal_store(v, (v4f*)(orow + (t + 128 * i) * 4));
    }
}

extern "C" void kernel_launch(void* const* d_in, const int* in_sizes, int n_in,
                              void* d_out, int out_size, void* d_ws, size_t ws_size,
                              hipStream_t stream) {
    const float* E    = (const float*)d_in[0];
    const float* Wq   = (const float*)d_in[1];
    const float* bq   = (const float*)d_in[2];
    const float* Wk   = (const float*)d_in[3];
    const float* bk   = (const float*)d_in[4];
    const int*   sidx = (const int*)  d_in[5];
    float*       out  = (float*)d_out;

    float* Qws = (float*)d_ws;                       // 4096x512 f32 = 8 MB
    float* Kws = Qws + (size_t)N_NODES * EMB;        // 4096x512 f32 = 8 MB

    dim3 gA(N_NODES / BM, EMB / BN, 2);              // 32 x 32 x 2 blocks
    proj_gemm_kernel<<<gA, 256, 0, stream>>>(E, Wq, bq, Wk, bk, Qws, Kws);

    sparse_row_kernel<<<HEADS * N_NODES, 128, 0, stream>>>(Qws, Kws, sidx, out);
}